// EWSCSQuantizer_61332132987278
// MI455X (gfx1250) — compile-verified
//
#include <hip/hip_runtime.h>
#include <hip/hip_bf16.h>

typedef __attribute__((ext_vector_type(2))) float v2f;
typedef __attribute__((ext_vector_type(8))) float v8f;

constexpr int kNPix  = 65536;   // 16 * 64 * 64 pixels
constexpr int kD     = 64;      // embedding dim
constexpr int kK     = 1024;    // num codes
constexpr int kHW    = 4096;    // 64*64 spatial per batch image
constexpr int kChunk = 128;     // codes staged in LDS per iteration
constexpr int kStr   = 68;      // padded LDS row stride (floats): bank-conflict free, 8B aligned

// ---------------------------------------------------------------------------
// Kernel 1: fused distance GEMM (f32 WMMA) + running argmin per pixel.
// Each wave owns 16 pixels (rows M of A). A = z tile 16x64 (f32), held in
// 16 v2f fragments. B = codebook tile 16 codes x 64 dims from LDS.
// score = z . e - 0.5*||e||^2  (C-matrix init); argmin dist == argmax score.
// ---------------------------------------------------------------------------
__global__ __launch_bounds__(256) void vq_argmin_wmma(
    const float* __restrict__ zg, const float* __restrict__ embg,
    unsigned short* __restrict__ idx_out) {
  __shared__ float lemb[kChunk * kStr];
  __shared__ float le2[kChunk];

  const int tid  = threadIdx.x;
  const int lane = tid & 31;
  const int wave = tid >> 5;
  const int m    = lane & 15;   // row within 16 (pixel / code within tile)
  const int kh   = lane >> 4;   // K-half select: lanes 0-15 K={0,1}, 16-31 K={2,3}

  const int pixel_base = blockIdx.x * 128 + wave * 16;
  const int p  = pixel_base + m;
  const int b  = p >> 12;              // 4096 pixels per batch image
  const int hw = p & (kHW - 1);
  const int zbase = b * (kD * kHW) + hw;  // z is [B, C, H, W]: channel stride = 4096

  // A fragments: 16 K-chunks of 4; per-lane 2 consecutive channels.
  v2f a[16];
#pragma unroll
  for (int kc = 0; kc < 16; ++kc) {
    const int c0 = kc * 4 + kh * 2;
    a[kc].x = zg[zbase + (c0 + 0) * kHW];
    a[kc].y = zg[zbase + (c0 + 1) * kHW];
  }

  float bs[8];
  int   bi[8];
#pragma unroll
  for (int r = 0; r < 8; ++r) { bs[r] = -3.4e38f; bi[r] = 0; }

  for (int ch = 0; ch < kK / kChunk; ++ch) {
    __syncthreads();
    // Stage 128 codes x 64 dims into LDS, coalesced global reads.
#pragma unroll
    for (int i = 0; i < (kChunk * kD) / 256; ++i) {
      const int lin  = i * 256 + tid;
      const int code = lin >> 6;
      const int d    = lin & 63;
      lemb[code * kStr + d] = embg[ch * (kChunk * kD) + lin];
    }
    if (ch + 1 < kK / kChunk)
      __builtin_prefetch(&embg[(ch + 1) * (kChunk * kD) + tid * 4], 0, 1);
    __syncthreads();
    if (tid < kChunk) {
      float s = 0.f;
#pragma unroll 8
      for (int d = 0; d < kD; ++d) {
        const float v = lemb[tid * kStr + d];
        s += v * v;
      }
      le2[tid] = s;
    }
    __syncthreads();

    for (int t = 0; t < kChunk / 16; ++t) {
      const int codeL   = t * 16 + m;
      const float cinit = -0.5f * le2[codeL];
      v8f acc = {cinit, cinit, cinit, cinit, cinit, cinit, cinit, cinit};
#pragma unroll
      for (int kc = 0; kc < 16; ++kc) {
        const v2f bb =
            *reinterpret_cast<const v2f*>(&lemb[codeL * kStr + kc * 4 + kh * 2]);
        acc = __builtin_amdgcn_wmma_f32_16x16x4_f32(
            false, a[kc], false, bb, (short)0, acc, false, false);
      }
      const int codeG = ch * kChunk + codeL;
#pragma unroll
      for (int r = 0; r < 8; ++r) {
        if (acc[r] > bs[r]) { bs[r] = acc[r]; bi[r] = codeG; }  // strict > keeps lowest idx
      }
    }
  }

  // Cross-lane reduction over the 16 columns (codes) of each half-wave.
  // Lanes 0-15 hold pixel M=r, lanes 16-31 hold pixel M=r+8.
#pragma unroll
  for (int off = 8; off >= 1; off >>= 1) {
#pragma unroll
    for (int r = 0; r < 8; ++r) {
      const float os = __shfl_xor(bs[r], off, 16);
      const int   oi = __shfl_xor(bi[r], off, 16);
      if (os > bs[r] || (os == bs[r] && oi < bi[r])) { bs[r] = os; bi[r] = oi; }
    }
  }

  if (m == 0) {
    const int rowBase = pixel_base + kh * 8;
#pragma unroll
    for (int r = 0; r < 8; ++r)
      idx_out[rowBase + r] = (unsigned short)bi[r];
  }
}

// ---------------------------------------------------------------------------
// Kernel 2: gather emb[idx] -> out (b,c,h,w), accumulate recon error + histogram
// ---------------------------------------------------------------------------
__global__ __launch_bounds__(256) void vq_gather(
    const float* __restrict__ zg, const float* __restrict__ embg,
    const unsigned short* __restrict__ idx, float* __restrict__ out,
    float* __restrict__ hist, float* __restrict__ err_sum) {
  __shared__ float red[256];
  const int p  = blockIdx.x * 256 + threadIdx.x;
  const int b  = p >> 12;
  const int hw = p & (kHW - 1);
  const int k  = idx[p];
  const int base = b * (kD * kHW) + hw;
  const float4* e4 = reinterpret_cast<const float4*>(embg + k * kD);
  float err = 0.f;
#pragma unroll
  for (int c4 = 0; c4 < kD / 4; ++c4) {
    const float4 e = e4[c4];
    const int c = c4 * 4;
    float zv, df;
    zv = zg[base + (c + 0) * kHW]; out[base + (c + 0) * kHW] = e.x; df = e.x - zv; err += df * df;
    zv = zg[base + (c + 1) * kHW]; out[base + (c + 1) * kHW] = e.y; df = e.y - zv; err += df * df;
    zv = zg[base + (c + 2) * kHW]; out[base + (c + 2) * kHW] = e.z; df = e.z - zv; err += df * df;
    zv = zg[base + (c + 3) * kHW]; out[base + (c + 3) * kHW] = e.w; df = e.w - zv; err += df * df;
  }
  atomicAdd(&hist[k], 1.0f);
  red[threadIdx.x] = err;
  __syncthreads();
  for (int s = 128; s >= 1; s >>= 1) {
    if (threadIdx.x < s) red[threadIdx.x] += red[threadIdx.x + s];
    __syncthreads();
  }
  if (threadIdx.x == 0) atomicAdd(err_sum, red[0]);
}

// ---------------------------------------------------------------------------
// Kernel 0: zero scratch accumulators (harness does not re-poison between runs)
// ---------------------------------------------------------------------------
__global__ void vq_init(float* __restrict__ hist, float* __restrict__ err_sum) {
  const int t = blockIdx.x * blockDim.x + threadIdx.x;
  if (t < kK) hist[t] = 0.f;
  if (t == 0) *err_sum = 0.f;
}

// ---------------------------------------------------------------------------
// Kernel 3: scalars -> loss, perplexity, avg_error
// ---------------------------------------------------------------------------
__global__ __launch_bounds__(1024) void vq_finalize(
    const float* __restrict__ hist, const float* __restrict__ err_sum,
    float* __restrict__ sc) {
  __shared__ float red[1024];
  const int t = threadIdx.x;
  const float c  = hist[t];
  const float pr = c * (1.0f / (float)kNPix);
  red[t] = -pr * logf(pr + 1e-10f);
  __syncthreads();
  for (int s = 512; s >= 1; s >>= 1) {
    if (t < s) red[t] += red[t + s];
    __syncthreads();
  }
  if (t == 0) {
    const float es = *err_sum;
    sc[0] = 1.25f * es / ((float)kNPix * (float)kD);  // q_loss + 0.25*e_loss (equal)
    sc[1] = expf(red[0]);                             // perplexity
    sc[2] = es / (float)kNPix;                        // avg per-pixel recon error
  }
}

extern "C" void kernel_launch(void* const* d_in, const int* in_sizes, int n_in,
                              void* d_out, int out_size, void* d_ws, size_t ws_size,
                              hipStream_t stream) {
  const float* z   = (const float*)d_in[0];   // [16, 64, 64, 64] f32
  const float* emb = (const float*)d_in[1];   // [1024, 64] f32
  float* out = (float*)d_out;                 // [4194304] out tensor + 3 scalars

  unsigned short* idx = (unsigned short*)d_ws;                       // 128 KB
  float* hist    = (float*)((char*)d_ws + kNPix * sizeof(unsigned short));
  float* err_sum = hist + kK;

  vq_init<<<(kK + 255) / 256, 256, 0, stream>>>(hist, err_sum);
  vq_argmin_wmma<<<kNPix / 128, 256, 0, stream>>>(z, emb, idx);
  vq_gather<<<kNPix / 256, 256, 0, stream>>>(z, emb, idx, out, hist, err_sum);
  vq_finalize<<<1, 1024, 0, stream>>>(hist, err_sum, out + kNPix * kD);
}